// HadamardModule_27659589386314
// MI455X (gfx1250) — compile-verified
//
#include <hip/hip_runtime.h>
#include <math.h>

// ---------------------------------------------------------------------------
// Fused: (x @ projector) -> [per-stack: d0*., FWHT128, d1*., FWHT128]
//        -> cos(C^2*z + 2*pi*b)   for MI455X (gfx1250, wave32, WMMA)
//
// Roofline: 256 MB output stores dominate (~11.7 us at 23.3 TB/s). GEMM uses
// V_WMMA_F32_16X16X4_F32 (full fp32 precision); FWHT in registers via wave32
// shfl_xor; cos via revolutions-domain v_cos_f32 with float-float reduction.
// ---------------------------------------------------------------------------

typedef __attribute__((ext_vector_type(2))) float v2f;
typedef __attribute__((ext_vector_type(4))) float v4f;
typedef __attribute__((ext_vector_type(8))) float v8f;

#define ROWS_PER_BLK 16
#define KDIM 512
#define NPCAS 128
#define KC 64                    // K chunk staged in LDS per iteration
#define NCHUNK (KDIM / KC)       // 8
#define NSTACKS 64
#define OUT_DIM 8192
#define XS_STRIDE 68             // 64 + 4 pad (stride % 64 banks == 4 -> conflict free)
#define PS_STRIDE 68             // projector stored transposed [n][k]
#define XT_STRIDE 132            // 128 + 4 pad for the 16x128 GEMM result tile

// COEFF_NORM^2 / (2*pi) = (128/9) / (2*pi), split into float-float
constexpr double CC_D =
    (128.0 / 9.0) / (2.0 * 3.14159265358979323846264338327950288);
#define CC_HI ((float)CC_D)
#define CC_LO ((float)(CC_D - (double)((float)CC_D)))

// length-128 FWHT, 4 contiguous elements per lane (col = 4*lane + j), wave32.
// stages h=1,2 intra-lane; h=4..64 cross-lane via shfl_xor(mask = h/4).
__device__ __forceinline__ void fwht128(float y[4], int lane) {
    float t;
    // h = 1
    t = y[0]; y[0] = t + y[1]; y[1] = t - y[1];
    t = y[2]; y[2] = t + y[3]; y[3] = t - y[3];
    // h = 2
    t = y[0]; y[0] = t + y[2]; y[2] = t - y[2];
    t = y[1]; y[1] = t + y[3]; y[3] = t - y[3];
    // h = 4,8,16,32,64 -> lane-xor 1,2,4,8,16
#pragma unroll
    for (int m = 1; m <= 16; m <<= 1) {
        const bool up = (lane & m) != 0;
#pragma unroll
        for (int j = 0; j < 4; ++j) {
            const float p = __shfl_xor(y[j], m, 32);
            y[j] = up ? (p - y[j]) : (y[j] + p);
        }
    }
}

__global__ __launch_bounds__(256) void had_fused_kernel(
    const float* __restrict__ xg,    // (8192, 512)
    const float* __restrict__ proj,  // (512, 128)
    const float* __restrict__ dmat,  // (2, 64, 128)
    const float* __restrict__ bvec,  // (8192,)
    float* __restrict__ out)         // (8192, 8192)
{
    extern __shared__ float lds[];
    float* xs = lds;                            // 16 x 68   x-tile K-chunk
    float* ps = xs + ROWS_PER_BLK * XS_STRIDE;  // 128 x 68  projector chunk (transposed)
    float* xt = ps + NPCAS * PS_STRIDE;         // 16 x 132  GEMM result tile

    const int t    = threadIdx.x;
    const int lane = t & 31;
    const int w    = t >> 5;        // wave 0..7
    const int n0   = w * 16;        // this wave's N-tile
    const int mn   = lane & 15;     // M (A / C) or N (B) index within tile
    const int kh   = lane >> 4;     // K half-select per ISA 32-bit fragment layout
    const long rowBase = (long)blockIdx.x * ROWS_PER_BLK;

    v8f acc = {0.f, 0.f, 0.f, 0.f, 0.f, 0.f, 0.f, 0.f};

    // ---------------- GEMM: 16 x 128 x 512 via V_WMMA_F32_16X16X4_F32 -------
    for (int c = 0; c < NCHUNK; ++c) {
        __syncthreads();
        // stage x tile chunk: 16 rows x 64 K (one NT float4 per thread;
        // x is streamed exactly once -> keep it out of L2's resident set)
        {
            const int r  = t >> 4;
            const int k4 = (t & 15) << 2;
            const v4f v = __builtin_nontemporal_load(
                reinterpret_cast<const v4f*>(&xg[(rowBase + r) * KDIM + c * KC + k4]));
            xs[r * XS_STRIDE + k4 + 0] = v.x;
            xs[r * XS_STRIDE + k4 + 1] = v.y;
            xs[r * XS_STRIDE + k4 + 2] = v.z;
            xs[r * XS_STRIDE + k4 + 3] = v.w;
        }
        // stage projector chunk transposed: ps[n][k] from proj[k][n] (64x128)
        // (projector is 256 KB total: L2-resident and reused by all blocks)
#pragma unroll
        for (int i = 0; i < 8; ++i) {
            const int v  = t + 256 * i;      // 0..2047
            const int kk = v >> 5;           // 0..63
            const int n4 = (v & 31) << 2;    // 0..124
            const v4f p = *reinterpret_cast<const v4f*>(
                &proj[(c * KC + kk) * NPCAS + n4]);
            ps[(n4 + 0) * PS_STRIDE + kk] = p.x;
            ps[(n4 + 1) * PS_STRIDE + kk] = p.y;
            ps[(n4 + 2) * PS_STRIDE + kk] = p.z;
            ps[(n4 + 3) * PS_STRIDE + kk] = p.w;
        }
        __syncthreads();
        // 16 WMMA K-steps of 4
#pragma unroll
        for (int kk = 0; kk < KC; kk += 4) {
            // A 16x4 f32: lanes0-15 = M, VGPR0:{K=kk+2kh}, VGPR1:{K=kk+2kh+1}
            const v2f a = *reinterpret_cast<const v2f*>(
                &xs[mn * XS_STRIDE + kk + 2 * kh]);
            // B 4x16 f32 (transposed store -> contiguous b64 per lane)
            const v2f bm = *reinterpret_cast<const v2f*>(
                &ps[(n0 + mn) * PS_STRIDE + kk + 2 * kh]);
            acc = __builtin_amdgcn_wmma_f32_16x16x4_f32(
                false, a, false, bm, (short)0, acc, false, false);
        }
    }

    // C tile 16x16 -> LDS: VGPR j holds (M = j + 8*kh, N = n0 + mn)
#pragma unroll
    for (int j = 0; j < 8; ++j) {
        xt[(j + 8 * kh) * XT_STRIDE + n0 + mn] = acc[j];
    }
    __syncthreads();

    // ---------------- per-stack sign * FWHT * cos, fully in registers -------
    // feats = C^2 * fwht(d1 * fwht(d0 * x_))  (COEFF folded out by linearity)
    // cos(feats + 2*pi*b) = v_cos( frac-reduced( feats/(2*pi) + b ) )
    float xv[2][4];
#pragma unroll
    for (int rr = 0; rr < 2; ++rr) {
        const v4f v = *reinterpret_cast<const v4f*>(
            &xt[(2 * w + rr) * XT_STRIDE + 4 * lane]);
        xv[rr][0] = v.x; xv[rr][1] = v.y; xv[rr][2] = v.z; xv[rr][3] = v.w;
    }
    const long rowg0 = rowBase + 2 * w;
    float* outb = out + rowg0 * OUT_DIM + 4 * lane;
    const float* dp0 = dmat + 4 * lane;
    const float* dp1 = dmat + NSTACKS * NPCAS + 4 * lane;
    const float* bp  = bvec + 4 * lane;

    for (int s = 0; s < NSTACKS; ++s) {
        const v4f d0 = *reinterpret_cast<const v4f*>(dp0 + s * NPCAS);
        const v4f d1 = *reinterpret_cast<const v4f*>(dp1 + s * NPCAS);
        const v4f bb = *reinterpret_cast<const v4f*>(bp + s * NPCAS);
        const float bbv[4] = { bb.x, bb.y, bb.z, bb.w };
#pragma unroll
        for (int rr = 0; rr < 2; ++rr) {
            float y[4] = { d0.x * xv[rr][0], d0.y * xv[rr][1],
                           d0.z * xv[rr][2], d0.w * xv[rr][3] };
            fwht128(y, lane);
            y[0] *= d1.x; y[1] *= d1.y; y[2] *= d1.z; y[3] *= d1.w;
            fwht128(y, lane);

            v4f o;
#pragma unroll
            for (int j = 0; j < 4; ++j) {
                // float-float: y*C^2/(2pi) in revolutions, then exact
                // integer-revolution strip (Sterbenz) + b; v_cos takes rev.
                const float hi = y[j] * CC_HI;
                float lo = fmaf(y[j], CC_HI, -hi);
                lo = fmaf(y[j], CC_LO, lo);
                const float rv = (hi - rintf(hi)) + lo + bbv[j];
                o[j] = __builtin_amdgcn_cosf(rv);
            }
            __builtin_nontemporal_store(
                o, reinterpret_cast<v4f*>(outb + (long)rr * OUT_DIM + s * NPCAS));
        }
    }
}

extern "C" void kernel_launch(void* const* d_in, const int* in_sizes, int n_in,
                              void* d_out, int out_size, void* d_ws, size_t ws_size,
                              hipStream_t stream) {
    (void)in_sizes; (void)n_in; (void)d_ws; (void)ws_size; (void)out_size;
    const float* x    = (const float*)d_in[0];
    const float* proj = (const float*)d_in[1];
    const float* d    = (const float*)d_in[2];
    const float* b    = (const float*)d_in[3];
    float* out = (float*)d_out;

    const dim3 grid(8192 / ROWS_PER_BLK);  // 512 blocks
    const dim3 block(256);                 // 8 wave32s
    const size_t shmem =
        (ROWS_PER_BLK * XS_STRIDE + NPCAS * PS_STRIDE + ROWS_PER_BLK * XT_STRIDE) *
        sizeof(float);                     // 47,616 B
    had_fused_kernel<<<grid, block, shmem, stream>>>(x, proj, d, b, out);
}